// GeometricLoss_42623255446154
// MI455X (gfx1250) — compile-verified
//
#include <hip/hip_runtime.h>
#include <hip/hip_bf16.h>

typedef __attribute__((ext_vector_type(2))) float v2f;
typedef __attribute__((ext_vector_type(8))) float v8f;

#define RDIM 256          // rows of each matrix (Gram is RDIM x RDIM)
#define CDIM 64           // cols (K dimension of the Gram matmul)
#define NMAT 2048         // 64 batches * 32 features
#define NB   16           // LU panel width (K of the trailing WMMA GEMM)
#define DET_OFF 1e-6f

#if __has_builtin(__builtin_amdgcn_wmma_f32_16x16x4_f32)
#define WMMA_F32X4(a, b, c) \
    __builtin_amdgcn_wmma_f32_16x16x4_f32(false, a, false, b, (short)0, c, false, false)
#else
// fallback (signals itself via wmma==0 in the histogram)
static __device__ inline v8f wmma_fallback(v2f a, v2f b, v8f c) {
    c[0] += a.x * b.x + a.y * b.y; return c;
}
#define WMMA_F32X4(a, b, c) wmma_fallback(a, b, c)
#endif

// LDS layout (dynamic, 327680 B = 320 KB, the full WGP budget):
//   [0     , 65536 ) : Pt[64][256] transposed P  (phase 1/2)
//                      -> reused in LU phase as:
//                         Pn[16][256]  panel, transposed  (floats 0..4095)
//                         diag[256]    U diagonal         (floats 4096..4351)
//                         piv_s[16]    pivot rows (int)   (floats 4352..4367)
//                         sstate[1]    swap parity (int)  (float  4384)
//                         pval/pidx    final reduction    (floats 0..511)
//   [65536 , 327680) : M[256][256]  Gram matrix / LU workspace

extern "C" __global__ void __launch_bounds__(256)
geomloss_main(const float* __restrict__ x, float* __restrict__ per_mat)
{
    extern __shared__ char smem_raw[];
    float* Pt = reinterpret_cast<float*>(smem_raw);      // 64*256 floats
    float* M  = Pt + CDIM * RDIM;                        // 256*256 floats

    float* Pn     = Pt;                                  // [NB][256]
    float* diag   = Pt + 4096;                           // [256]
    int*   piv_s  = reinterpret_cast<int*>(Pt + 4352);   // [NB]
    int*   sstate = reinterpret_cast<int*>(Pt + 4384);   // [1]

    const int tid   = threadIdx.x;                       // 0..255 (8 wave32)
    const int wave  = tid >> 5;
    const int lane  = tid & 31;
    const int lhalf = lane >> 4;                         // 0 / 1
    const int lmod  = lane & 15;                         // 0..15
    const int bf    = blockIdx.x;                        // 0..2047
    const float* __restrict__ xm = x + (size_t)bf * (RDIM * CDIM);

    // ---------------- Phase 1: load + transpose + min/max + normalize ----
    float mn =  __builtin_inff();
    float mx = -__builtin_inff();
    #pragma unroll 4
    for (int i = 0; i < 64; ++i) {
        const int e = i * 256 + tid;     // coalesced global reads
        const int r = e >> 6;
        const int c = e & 63;
        const float v = xm[e];
        Pt[c * RDIM + r] = v;            // store transposed
        mn = fminf(mn, v);
        mx = fmaxf(mx, v);
    }
    {   // block min/max tree reduction in the (idle) M region
        float* smn = M;
        float* smx = M + 256;
        smn[tid] = mn; smx[tid] = mx;
        __syncthreads();
        for (int s = 128; s > 0; s >>= 1) {
            if (tid < s) {
                smn[tid] = fminf(smn[tid], smn[tid + s]);
                smx[tid] = fmaxf(smx[tid], smx[tid + s]);
            }
            __syncthreads();
        }
        mn = smn[0]; mx = smx[0];
        __syncthreads();
    }
    const bool  valid = (mx > mn);
    const float inv   = 1.0f / (valid ? (mx - mn) : 1.0f);
    #pragma unroll 4
    for (int i = 0; i < 64; ++i) {
        const int idx = i * 256 + tid;
        Pt[idx] = (Pt[idx] - mn) * inv;
    }
    __syncthreads();

    // ---------------- Phase 2: Gram matrix via fp32 WMMA -----------------
    for (int t = wave; t < 256; t += 8) {            // wave-uniform tiles
        const int i0 = (t >> 4) * 16;
        const int j0 = (t & 15) * 16;
        v8f acc = {};
        #pragma unroll
        for (int k0 = 0; k0 < CDIM; k0 += 4) {
            const int ka = k0 + 2 * lhalf;
            v2f a, b;
            a.x = Pt[(ka    ) * RDIM + i0 + lmod];
            a.y = Pt[(ka + 1) * RDIM + i0 + lmod];
            b.x = Pt[(ka    ) * RDIM + j0 + lmod];
            b.y = Pt[(ka + 1) * RDIM + j0 + lmod];
            acc = WMMA_F32X4(a, b, acc);
        }
        #pragma unroll
        for (int v = 0; v < 8; ++v)
            M[(i0 + v + 8 * lhalf) * RDIM + (j0 + lmod)] = acc[v] + DET_OFF;
    }
    __syncthreads();

    // ---------------- Phase 3: blocked LU (NB=16) with partial pivoting --
    if (tid == 0) sstate[0] = 0;

    for (int pb = 0; pb < RDIM / NB; ++pb) {
        const int j0 = pb * NB;
        const int W  = RDIM - j0 - NB;               // trailing width

        // ---- copy panel rows >= j0 into Pn, transposed ----
        {
            const int i = j0 + tid;
            if (i < RDIM) {
                #pragma unroll
                for (int c = 0; c < NB; ++c)
                    Pn[c * RDIM + i] = M[i * RDIM + (j0 + c)];
            }
        }
        __syncthreads();

        // ---- panel factorization: wave 0 only, zero block barriers ----
        if (wave == 0) {
            int parity = 0;
            for (int cl = 0; cl < NB; ++cl) {
                const int jj = j0 + cl;
                // pivot argmax over rows i >= jj (conflict-free row reads)
                float bv = -1.0f; int bi = jj;
                #pragma unroll
                for (int w = 0; w < 8; ++w) {
                    const int i = lane + 32 * w;
                    if (i >= jj) {
                        const float a = fabsf(Pn[cl * RDIM + i]);
                        if (a > bv) { bv = a; bi = i; }
                    }
                }
                #pragma unroll
                for (int m = 16; m >= 1; m >>= 1) {
                    const float ov = __shfl_xor(bv, m, 32);
                    const int   oi = __shfl_xor(bi, m, 32);
                    if (ov > bv || (ov == bv && oi < bi)) { bv = ov; bi = oi; }
                }
                const int pr = bi;                    // uniform across wave
                if (lane == 0) piv_s[cl] = pr;
                if (pr != jj) {                       // swap panel rows jj<->pr
                    parity ^= 1;
                    if (lane < NB) {
                        const float t0 = Pn[lane * RDIM + jj];
                        const float t1 = Pn[lane * RDIM + pr];
                        Pn[lane * RDIM + jj] = t1;
                        Pn[lane * RDIM + pr] = t0;
                    }
                }
                asm volatile("s_wait_dscnt 0" ::: "memory");
                const float pvv = Pn[cl * RDIM + jj];
                // scale column below pivot; keep L values in registers
                float lv[8];
                #pragma unroll
                for (int w = 0; w < 8; ++w) {
                    const int i = lane + 32 * w;
                    float l = 0.0f;
                    if (i > jj && pvv != 0.0f) {
                        l = Pn[cl * RDIM + i] / pvv;
                        Pn[cl * RDIM + i] = l;
                    }
                    lv[w] = l;
                }
                asm volatile("s_wait_dscnt 0" ::: "memory");
                // rank-1 update of remaining panel columns
                for (int c2 = cl + 1; c2 < NB; ++c2) {
                    const float u = Pn[c2 * RDIM + jj];
                    #pragma unroll
                    for (int w = 0; w < 8; ++w) {
                        const int i = lane + 32 * w;
                        if (i > jj)
                            Pn[c2 * RDIM + i] -= lv[w] * u;
                    }
                }
                asm volatile("s_wait_dscnt 0" ::: "memory");
            }
            if (lane == 0) sstate[0] ^= parity;
        }
        __syncthreads();

        // ---- bank the U diagonal of this panel ----
        if (tid < NB) diag[j0 + tid] = Pn[tid * RDIM + (j0 + tid)];

        // ---- deferred row swaps on trailing cols + U12 forward solve ----
        // (each thread owns one trailing column -> no internal barriers)
        if (tid < W) {
            const int c = j0 + NB + tid;
            #pragma unroll
            for (int k = 0; k < NB; ++k) {
                const int r = j0 + k;
                const int q = piv_s[k];
                if (q != r) {
                    const float t0 = M[r * RDIM + c];
                    M[r * RDIM + c] = M[q * RDIM + c];
                    M[q * RDIM + c] = t0;
                }
            }
            float u[NB];
            #pragma unroll
            for (int k = 0; k < NB; ++k) u[k] = M[(j0 + k) * RDIM + c];
            #pragma unroll
            for (int k = 1; k < NB; ++k) {
                float s = u[k];
                for (int m2 = 0; m2 < k; ++m2)
                    s -= Pn[m2 * RDIM + (j0 + k)] * u[m2];  // L11 broadcast
                u[k] = s;
            }
            #pragma unroll
            for (int k = 0; k < NB; ++k) M[(j0 + k) * RDIM + c] = u[k];
        }
        __syncthreads();

        // ---- trailing update A22 -= L21 * U12 via WMMA (K = 16) ----
        const int T = W >> 4;                        // trailing tiles per dim
        for (int t = wave; t < T * T; t += 8) {      // wave-uniform tiles
            const int i0 = j0 + NB + (t / T) * 16;
            const int c0 = j0 + NB + (t % T) * 16;
            v8f acc = {};
            #pragma unroll
            for (int k0 = 0; k0 < NB; k0 += 4) {
                const int ka = k0 + 2 * lhalf;
                v2f a, b;
                a.x = Pn[(ka    ) * RDIM + i0 + lmod];       // L21 (transposed)
                a.y = Pn[(ka + 1) * RDIM + i0 + lmod];
                b.x = M[(j0 + ka    ) * RDIM + c0 + lmod];   // U12 rows
                b.y = M[(j0 + ka + 1) * RDIM + c0 + lmod];
                acc = WMMA_F32X4(a, b, acc);
            }
            #pragma unroll
            for (int v = 0; v < 8; ++v) {
                const int idx = (i0 + v + 8 * lhalf) * RDIM + (c0 + lmod);
                M[idx] -= acc[v];
            }
        }
        __syncthreads();
    }

    // ---------------- Phase 4: logdet from banked U diagonal -------------
    {
        float* pval = Pt;                                // reuse Pn region
        int*   pidx = reinterpret_cast<int*>(Pt + 256);
        const float d = diag[tid];
        pval[tid] = logf(fabsf(d));
        pidx[tid] = (d < 0.0f ? 1 : 0) | (d == 0.0f ? 2 : 0);
        __syncthreads();
        for (int s = 128; s > 0; s >>= 1) {
            if (tid < s) {
                pval[tid] += pval[tid + s];
                const int a = pidx[tid], b = pidx[tid + s];
                pidx[tid] = ((a ^ b) & 1) | ((a | b) & 2);
            }
            __syncthreads();
        }
        if (tid == 0) {
            const float logabs  = pval[0];
            const int   negpar  = (pidx[0] & 1) ^ (sstate[0] & 1);
            const bool  anyzero = (pidx[0] & 2) != 0;
            const float sign    = anyzero ? 0.0f : (negpar ? -1.0f : 1.0f);
            float logdet;
            if (sign > 0.0f)      logdet = logabs;
            else if (sign == 0.f) logdet = -__builtin_inff();
            else                  logdet = __builtin_nanf("");
            const bool keep = valid && __builtin_isfinite(logdet);
            per_mat[bf] = keep ? fabsf(logdet) : __builtin_nanf("");
        }
    }
}

// Deterministic final reduction: mean of non-NaN entries.
extern "C" __global__ void __launch_bounds__(256)
geomloss_reduce(const float* __restrict__ per_mat, float* __restrict__ out)
{
    __shared__ float ssum[256];
    __shared__ float scnt[256];
    const int tid = threadIdx.x;
    float s = 0.0f, c = 0.0f;
    for (int i = tid; i < NMAT; i += 256) {
        const float v = per_mat[i];
        if (v == v) { s += v; c += 1.0f; }   // NaN-safe keep test
    }
    ssum[tid] = s; scnt[tid] = c;
    __syncthreads();
    for (int st = 128; st > 0; st >>= 1) {
        if (tid < st) {
            ssum[tid] += ssum[tid + st];
            scnt[tid] += scnt[tid + st];
        }
        __syncthreads();
    }
    if (tid == 0) out[0] = ssum[0] / scnt[0];
}

extern "C" void kernel_launch(void* const* d_in, const int* in_sizes, int n_in,
                              void* d_out, int out_size, void* d_ws, size_t ws_size,
                              hipStream_t stream) {
    const float* x  = (const float*)d_in[0];
    float* out      = (float*)d_out;
    float* per_mat  = (float*)d_ws;                  // NMAT floats of scratch
    (void)in_sizes; (void)n_in; (void)out_size; (void)ws_size;

    const size_t lds_bytes = (size_t)(CDIM * RDIM + RDIM * RDIM) * sizeof(float); // 320 KB
    geomloss_main<<<NMAT, 256, lds_bytes, stream>>>(x, per_mat);
    geomloss_reduce<<<1, 256, 0, stream>>>(per_mat, out);
}